// AttentionDecoderRNN_32392643346983
// MI455X (gfx1250) — compile-verified
//
#include <hip/hip_runtime.h>
#include <math.h>

#define H 1024
#define V 50257
#define L 512

typedef float v2f __attribute__((ext_vector_type(2)));
typedef float v8f __attribute__((ext_vector_type(8)));

// ---------------------------------------------------------------------------
// K1: attn_logits[r] = dot(W_attn[r,:], concat(emb_row, h)) + b_attn[r]
// one wave32 per row; 512 rows; coalesced float4 loads
// ---------------------------------------------------------------------------
__global__ void k_attn_logits(const float* __restrict__ Wa, const float* __restrict__ ba,
                              const float* __restrict__ emb, const int* __restrict__ inp,
                              const float* __restrict__ hid, float* __restrict__ logits) {
    __shared__ alignas(16) float a2[2 * H];
    int idx = inp[0];
    for (int k = threadIdx.x; k < 2 * H; k += 256)
        a2[k] = (k < H) ? emb[(size_t)idx * H + k] : hid[k - H];
    __syncthreads();
    int row  = blockIdx.x * 8 + (threadIdx.x >> 5);
    int lane = threadIdx.x & 31;
    const float4* wp = (const float4*)(Wa + (size_t)row * (2 * H));
    const float4* ap = (const float4*)a2;
    float acc = 0.f;
#pragma unroll
    for (int i = 0; i < 16; ++i) {
        float4 w = wp[lane + 32 * i];
        float4 a = ap[lane + 32 * i];
        acc = fmaf(w.x, a.x, fmaf(w.y, a.y, fmaf(w.z, a.z, fmaf(w.w, a.w, acc))));
    }
    for (int off = 16; off; off >>= 1) acc += __shfl_xor(acc, off, 32);
    if (lane == 0) logits[row] = acc + ba[row];
}

// ---------------------------------------------------------------------------
// K2: softmax over 512 logits -> attn_weights (written straight into d_out)
// ---------------------------------------------------------------------------
__global__ void k_softmax(const float* __restrict__ logits, float* __restrict__ weights) {
    __shared__ float red[L];
    int t = threadIdx.x;
    float v = logits[t];
    red[t] = v; __syncthreads();
    for (int s = 256; s; s >>= 1) { if (t < s) red[t] = fmaxf(red[t], red[t + s]); __syncthreads(); }
    float mx = red[0]; __syncthreads();
    float e = expf(v - mx);
    red[t] = e; __syncthreads();
    for (int s = 256; s; s >>= 1) { if (t < s) red[t] += red[t + s]; __syncthreads(); }
    weights[t] = e / red[0];
}

// ---------------------------------------------------------------------------
// K3a/K3b: attn_applied[h] = sum_l w[l] * enc[l,h]   (2-stage, deterministic)
// ---------------------------------------------------------------------------
__global__ void k_apply_partial(const float* __restrict__ w, const float* __restrict__ enc,
                                float* __restrict__ partial) {
    int c = blockIdx.x;                       // 8 chunks of 64 rows
    for (int h = threadIdx.x; h < H; h += 256) {
        float s = 0.f;
        for (int l = c * 64; l < c * 64 + 64; ++l)
            s = fmaf(w[l], enc[(size_t)l * H + h], s);
        partial[c * H + h] = s;
    }
}

__global__ void k_apply_reduce(const float* __restrict__ partial, float* __restrict__ applied) {
    int h = blockIdx.x * 256 + threadIdx.x;
    float s = 0.f;
#pragma unroll
    for (int c = 0; c < 8; ++c) s += partial[c * H + h];
    applied[h] = s;
}

// ---------------------------------------------------------------------------
// K4: x = relu(W_comb @ concat(emb_row, attn_applied) + b_comb)   [WMMA f32]
// one wave per 16 output rows; V_WMMA_F32_16X16X4_F32, B broadcast over N
// ---------------------------------------------------------------------------
__global__ void k_combine_wmma(const float* __restrict__ Wc, const float* __restrict__ bc,
                               const float* __restrict__ emb, const int* __restrict__ inp,
                               const float* __restrict__ applied, float* __restrict__ xout) {
    __shared__ alignas(16) float a2[2 * H];
    int idx = inp[0];
    for (int k = threadIdx.x; k < 2 * H; k += 256)
        a2[k] = (k < H) ? emb[(size_t)idx * H + k] : applied[k - H];
    __syncthreads();
    int wave = (blockIdx.x * 256 + threadIdx.x) >> 5;   // 0..63 row-tiles
    int lane = threadIdx.x & 31;
    int mrow = (wave << 4) + (lane & 15);               // A: lane%16 -> M
    int ko   = (lane >> 4) << 1;                        // lanes>=16 hold K+2
    const float* ap = Wc + (size_t)mrow * (2 * H);
    v8f c = {};
    for (int kb = 0; kb < 2 * H; kb += 4) {
        int kk = kb + ko;
        v2f a = *(const v2f*)(ap + kk);                 // A[M, kk..kk+1]
        v2f b = *(const v2f*)(&a2[kk]);                 // B[kk..kk+1, *] broadcast over N
        c = __builtin_amdgcn_wmma_f32_16x16x4_f32(false, a, false, b, (short)0, c, false, false);
    }
    if ((lane & 15) == 0) {                             // lanes 0,16 hold column N=0
        int mb = (wave << 4) + ((lane >> 4) << 3);
#pragma unroll
        for (int j = 0; j < 8; ++j) {
            float vv = c[j] + bc[mb + j];
            xout[mb + j] = vv > 0.f ? vv : 0.f;
        }
    }
}

// ---------------------------------------------------------------------------
// K5: gi = W_ih @ x + b_ih ; gh = W_hh @ h + b_hh    [WMMA f32, dual accum]
// 192 row-tiles of 16 over 3H outputs
// ---------------------------------------------------------------------------
__global__ void k_gates_wmma(const float* __restrict__ Wih, const float* __restrict__ Whh,
                             const float* __restrict__ bih, const float* __restrict__ bhh,
                             const float* __restrict__ xv, const float* __restrict__ hv,
                             float* __restrict__ gi, float* __restrict__ gh) {
    __shared__ alignas(16) float xs[H];
    __shared__ alignas(16) float hs[H];
    for (int k = threadIdx.x; k < H; k += 256) { xs[k] = xv[k]; hs[k] = hv[k]; }
    __syncthreads();
    int wave = (blockIdx.x * 256 + threadIdx.x) >> 5;   // 0..191
    int lane = threadIdx.x & 31;
    int mrow = (wave << 4) + (lane & 15);
    int ko   = (lane >> 4) << 1;
    const float* api = Wih + (size_t)mrow * H;
    const float* aph = Whh + (size_t)mrow * H;
    v8f ci = {}, ch = {};
    for (int kb = 0; kb < H; kb += 4) {
        int kk = kb + ko;
        v2f ai = *(const v2f*)(api + kk);
        v2f ah = *(const v2f*)(aph + kk);
        v2f bx = *(const v2f*)(&xs[kk]);
        v2f bh = *(const v2f*)(&hs[kk]);
        ci = __builtin_amdgcn_wmma_f32_16x16x4_f32(false, ai, false, bx, (short)0, ci, false, false);
        ch = __builtin_amdgcn_wmma_f32_16x16x4_f32(false, ah, false, bh, (short)0, ch, false, false);
    }
    if ((lane & 15) == 0) {
        int mb = (wave << 4) + ((lane >> 4) << 3);
#pragma unroll
        for (int j = 0; j < 8; ++j) {
            gi[mb + j] = ci[j] + bih[mb + j];
            gh[mb + j] = ch[j] + bhh[mb + j];
        }
    }
}

// ---------------------------------------------------------------------------
// K6: GRU gate math -> h_new (written into d_out[V..V+H))
// ---------------------------------------------------------------------------
__global__ void k_gru(const float* __restrict__ gi, const float* __restrict__ gh,
                      const float* __restrict__ hid, float* __restrict__ hnew) {
    int j = blockIdx.x * 256 + threadIdx.x;
    float r = 1.f / (1.f + expf(-(gi[j] + gh[j])));
    float z = 1.f / (1.f + expf(-(gi[H + j] + gh[H + j])));
    float n = tanhf(gi[2 * H + j] + r * gh[2 * H + j]);
    hnew[j] = (1.f - z) * n + z * hid[j];
}

// ---------------------------------------------------------------------------
// K7: logits[v] = dot(W_out[v,:], h_new) + b_out[v]  -- THE 206MB stream.
// One wave32 per row, float4 coalesced, LDS-staged h_new.
// ---------------------------------------------------------------------------
__global__ void k_out_gemv(const float* __restrict__ Wout, const float* __restrict__ bout,
                           const float* __restrict__ hn, float* __restrict__ out) {
    __shared__ alignas(16) float hsh[H];
    for (int k = threadIdx.x; k < H; k += 256) hsh[k] = hn[k];
    __syncthreads();
    int row  = blockIdx.x * 8 + (threadIdx.x >> 5);
    int lane = threadIdx.x & 31;
    if (row < V) {
        const float4* wp = (const float4*)(Wout + (size_t)row * H);
        const float4* hp = (const float4*)hsh;
        float acc = 0.f;
#pragma unroll
        for (int i = 0; i < 8; ++i) {
            float4 w = wp[lane + 32 * i];
            float4 h = hp[lane + 32 * i];
            acc = fmaf(w.x, h.x, fmaf(w.y, h.y, fmaf(w.z, h.z, fmaf(w.w, h.w, acc))));
        }
        for (int off = 16; off; off >>= 1) acc += __shfl_xor(acc, off, 32);
        if (lane == 0) out[row] = acc + bout[row];
    }
}

// ---------------------------------------------------------------------------
// K8a/K8b: online logsumexp over V logits (deterministic 2-stage)
// ---------------------------------------------------------------------------
__global__ void k_lse_partial(const float* __restrict__ logits, float* __restrict__ pairs) {
    int t = blockIdx.x * 256 + threadIdx.x;             // 64*256 = 16384 threads
    float m = -INFINITY, s = 0.f;
    for (int v = t; v < V; v += 64 * 256) {
        float l = logits[v];
        if (l > m) { s = s * expf(m - l) + 1.f; m = l; }
        else       { s += expf(l - m); }
    }
    __shared__ float sm[256], ss[256];
    sm[threadIdx.x] = m; ss[threadIdx.x] = s; __syncthreads();
    for (int st = 128; st; st >>= 1) {
        if (threadIdx.x < st) {
            float m1 = sm[threadIdx.x], s1 = ss[threadIdx.x];
            float m2 = sm[threadIdx.x + st], s2 = ss[threadIdx.x + st];
            float M = fmaxf(m1, m2);
            sm[threadIdx.x] = M;
            ss[threadIdx.x] = s1 * expf(m1 - M) + s2 * expf(m2 - M);
        }
        __syncthreads();
    }
    if (threadIdx.x == 0) { pairs[blockIdx.x * 2] = sm[0]; pairs[blockIdx.x * 2 + 1] = ss[0]; }
}

__global__ void k_lse_final(const float* __restrict__ pairs, float* __restrict__ cval) {
    __shared__ float sm[64], ss[64];
    int t = threadIdx.x;
    sm[t] = pairs[2 * t]; ss[t] = pairs[2 * t + 1]; __syncthreads();
    for (int st = 32; st; st >>= 1) {
        if (t < st) {
            float m1 = sm[t], s1 = ss[t];
            float m2 = sm[t + st], s2 = ss[t + st];
            float M = fmaxf(m1, m2);
            sm[t] = M;
            ss[t] = s1 * expf(m1 - M) + s2 * expf(m2 - M);
        }
        __syncthreads();
    }
    if (t == 0) cval[0] = sm[0] + logf(ss[0]);
}

__global__ void k_lsm_finalize(float* __restrict__ out, const float* __restrict__ cval) {
    int v = blockIdx.x * 256 + threadIdx.x;
    if (v < V) out[v] -= cval[0];
}

// ---------------------------------------------------------------------------
extern "C" void kernel_launch(void* const* d_in, const int* in_sizes, int n_in,
                              void* d_out, int out_size, void* d_ws, size_t ws_size,
                              hipStream_t stream) {
    const int*   inp   = (const int*)  d_in[0];
    const float* hid   = (const float*)d_in[1];
    const float* enc   = (const float*)d_in[2];
    const float* emb   = (const float*)d_in[3];
    const float* Wa    = (const float*)d_in[4];
    const float* ba    = (const float*)d_in[5];
    const float* Wc    = (const float*)d_in[6];
    const float* bc    = (const float*)d_in[7];
    const float* Wih   = (const float*)d_in[8];
    const float* Whh   = (const float*)d_in[9];
    const float* bih   = (const float*)d_in[10];
    const float* bhh   = (const float*)d_in[11];
    const float* Wout  = (const float*)d_in[12];
    const float* bout  = (const float*)d_in[13];

    float* out    = (float*)d_out;        // [V] log_softmax
    float* hnew   = out + V;              // [H] h_new
    float* attw   = out + V + H;          // [L] attn_weights

    float* ws       = (float*)d_ws;
    float* w_logits = ws;                 // 512
    float* w_part   = ws + 512;           // 8*1024
    float* w_appl   = ws + 512 + 8 * H;   // 1024
    float* w_x      = w_appl + H;         // 1024
    float* w_gi     = w_x + H;            // 3072
    float* w_gh     = w_gi + 3 * H;       // 3072
    float* w_pairs  = w_gh + 3 * H;       // 128
    float* w_c      = w_pairs + 128;      // 1

    k_attn_logits  <<<64,   256, 0, stream>>>(Wa, ba, emb, inp, hid, w_logits);
    k_softmax      <<<1,    512, 0, stream>>>(w_logits, attw);
    k_apply_partial<<<8,    256, 0, stream>>>(attw, enc, w_part);
    k_apply_reduce <<<4,    256, 0, stream>>>(w_part, w_appl);
    k_combine_wmma <<<8,    256, 0, stream>>>(Wc, bc, emb, inp, w_appl, w_x);
    k_gates_wmma   <<<24,   256, 0, stream>>>(Wih, Whh, bih, bhh, w_x, hid, w_gi, w_gh);
    k_gru          <<<4,    256, 0, stream>>>(w_gi, w_gh, hid, hnew);
    k_out_gemv     <<<(V + 7) / 8, 256, 0, stream>>>(Wout, bout, hnew, out);
    k_lse_partial  <<<64,   256, 0, stream>>>(out, w_pairs);
    k_lse_final    <<<1,     64, 0, stream>>>(w_pairs, w_c);
    k_lsm_finalize <<<(V + 255) / 256, 256, 0, stream>>>(out, w_c);
}